// MeshLoss_72730976191061
// MI455X (gfx1250) — compile-verified
//
#include <hip/hip_runtime.h>
#include <hip/hip_bf16.h>

#define EPS 1e-12f
#define CHAMFER_W 1.0f
#define NORM_W 0.1f
#define EDGE_W 0.5f

typedef float v2f __attribute__((ext_vector_type(2)));
typedef float v8f __attribute__((ext_vector_type(8)));

// ---------------------------------------------------------------------------
// Kernel 1: per-face normals and areas.  grid-stride over B*F.
// ---------------------------------------------------------------------------
__global__ void face_kernel(const float* __restrict__ verts,   // [B,V,3]
                            const int* __restrict__ faces,     // [F,3]
                            float* __restrict__ normals,       // [B,F,3]
                            float* __restrict__ areas,         // [B,F]
                            int B, int V, int F) {
    int total = B * F;
    for (int i = blockIdx.x * blockDim.x + threadIdx.x; i < total;
         i += gridDim.x * blockDim.x) {
        int b = i / F, f = i - b * F;
        int i0 = faces[3 * f + 0], i1 = faces[3 * f + 1], i2 = faces[3 * f + 2];
        const float* vb = verts + (size_t)b * V * 3;
        float ax = vb[3 * i0], ay = vb[3 * i0 + 1], az = vb[3 * i0 + 2];
        float bx = vb[3 * i1], by = vb[3 * i1 + 1], bz = vb[3 * i1 + 2];
        float cx = vb[3 * i2], cy = vb[3 * i2 + 1], cz = vb[3 * i2 + 2];
        float e1x = bx - ax, e1y = by - ay, e1z = bz - az;
        float e2x = cx - ax, e2y = cy - ay, e2z = cz - az;
        float nx = e1y * e2z - e1z * e2y;
        float ny = e1z * e2x - e1x * e2z;
        float nz = e1x * e2y - e1y * e2x;
        float cn = sqrtf(nx * nx + ny * ny + nz * nz);
        areas[i] = 0.5f * cn;
        float inv = 1.0f / (cn + EPS);
        normals[(size_t)i * 3 + 0] = nx * inv;
        normals[(size_t)i * 3 + 1] = ny * inv;
        normals[(size_t)i * 3 + 2] = nz * inv;
    }
}

// ---------------------------------------------------------------------------
// Kernel 2: per-batch CDF scan (in LDS) + searchsorted + barycentric sample.
// One 1024-thread block per batch.  Handles both rand sets (pred & gt).
// Writes K-padded transposed points [B,4,N] for WMMA operand loads.
// ---------------------------------------------------------------------------
__global__ void sample_kernel(const float* __restrict__ verts,   // [B,V,3]
                              const int* __restrict__ faces,     // [F,3]
                              const float* __restrict__ normals, // [B,F,3]
                              const float* __restrict__ areas,   // [B,F]
                              const float* __restrict__ rnd_p,   // [B,N,3]
                              const float* __restrict__ rnd_g,   // [B,N,3]
                              float* __restrict__ pcT,           // [B,4,N]
                              float* __restrict__ gcT,           // [B,4,N]
                              float* __restrict__ pn,            // [B,N,3]
                              float* __restrict__ gn,            // [B,N,3]
                              float* __restrict__ pn2,           // [B,N]
                              float* __restrict__ gn2,           // [B,N]
                              int B, int V, int F, int N) {
    extern __shared__ float smem[];
    float* cdf = smem;          // F floats
    float* wp  = smem + F;      // 32 warp partials

    int b    = blockIdx.x;
    int tid  = threadIdx.x;
    int lane = tid & 31;
    int warp = tid >> 5;
    int nthr = blockDim.x;
    const float* ab = areas + (size_t)b * F;

    // --- per-thread sequential chunk sum ---
    int chunk = (F + nthr - 1) / nthr;
    int s0 = tid * chunk;
    int s1 = min(F, s0 + chunk);
    float sum = 0.0f;
    for (int i = s0; i < s1; ++i) sum += ab[i];

    // --- wave32 inclusive scan ---
    float v = sum;
    #pragma unroll
    for (int off = 1; off < 32; off <<= 1) {
        float n = __shfl_up(v, off, 32);
        if (lane >= off) v += n;
    }
    if (lane == 31) wp[warp] = v;
    __syncthreads();
    if (warp == 0) {
        float w = wp[lane];
        #pragma unroll
        for (int off = 1; off < 32; off <<= 1) {
            float n = __shfl_up(w, off, 32);
            if (lane >= off) w += n;
        }
        wp[lane] = w;  // inclusive scan of warp sums
    }
    __syncthreads();
    float excl = (v - sum) + (warp > 0 ? wp[warp - 1] : 0.0f);

    // --- fill inclusive cdf ---
    float run = excl;
    for (int i = s0; i < s1; ++i) { run += ab[i]; cdf[i] = run; }
    __syncthreads();
    float inv = 1.0f / (cdf[F - 1] + EPS);
    for (int i = s0; i < s1; ++i) cdf[i] *= inv;
    __syncthreads();

    // --- sample 2*N points ---
    for (int s = tid; s < 2 * N; s += nthr) {
        bool isP = s < N;
        int n = isP ? s : s - N;
        const float* r = (isP ? rnd_p : rnd_g) + ((size_t)b * N + n) * 3;
        float u = r[0], r1 = r[1], r2 = r[2];
        // searchsorted (side='left'): first idx with cdf[idx] >= u
        int lo = 0, hi = F;
        while (lo < hi) {
            int mid = (lo + hi) >> 1;
            if (cdf[mid] < u) lo = mid + 1; else hi = mid;
        }
        int idx = min(lo, F - 1);
        int i0 = faces[3 * idx], i1 = faces[3 * idx + 1], i2 = faces[3 * idx + 2];
        const float* vb = verts + (size_t)b * V * 3;
        float su = sqrtf(r1);
        float w0 = 1.0f - su, w1 = su * (1.0f - r2), w2 = su * r2;
        float px = w0 * vb[3 * i0 + 0] + w1 * vb[3 * i1 + 0] + w2 * vb[3 * i2 + 0];
        float py = w0 * vb[3 * i0 + 1] + w1 * vb[3 * i1 + 1] + w2 * vb[3 * i2 + 1];
        float pz = w0 * vb[3 * i0 + 2] + w1 * vb[3 * i1 + 2] + w2 * vb[3 * i2 + 2];
        float* T = (isP ? pcT : gcT) + (size_t)b * 4 * N;
        T[0 * N + n] = px;
        T[1 * N + n] = py;
        T[2 * N + n] = pz;
        T[3 * N + n] = 0.0f;   // K-pad
        const float* nrm = normals + ((size_t)b * F + idx) * 3;
        float* NO = (isP ? pn : gn) + ((size_t)b * N + n) * 3;
        NO[0] = nrm[0]; NO[1] = nrm[1]; NO[2] = nrm[2];
        (isP ? pn2 : gn2)[(size_t)b * N + n] = px * px + py * py + pz * pz;
    }
}

// ---------------------------------------------------------------------------
// Kernel 3: fused chamfer min/argmin via V_WMMA_F32_16X16X4_F32.
// One wave per 16-row tile of A; loops over all 16-col tiles of B,
// 4 tiles per trip: 12 clause-batched loads feed 4 back-to-back WMMAs.
// Never materializes the distance matrix (saves ~537 MB of traffic).
// A operand (16x4 f32): lanes 0-15 hold M=lane, K={0,1}; lanes 16-31 K={2,3}.
// B operand (4x16 f32): lanes 0-15 hold N=lane, K={0,1}; lanes 16-31 K={2,3}.
// C/D (16x16 f32): VGPR j <-> M = j + 8*(lane>=16); N = lane&15.
// ---------------------------------------------------------------------------
__global__ void chamfer_kernel(const float* __restrict__ AT,   // [B,4,N]
                               const float* __restrict__ BT,   // [B,4,N]
                               const float* __restrict__ an2,  // [B,N]
                               const float* __restrict__ bn2,  // [B,N]
                               float* __restrict__ dist_out,   // [B,N]
                               int* __restrict__ nn_out,       // [B,N] or null
                               int B, int N) {
    int lane = threadIdx.x & 31;
    int wave = (blockIdx.x * blockDim.x + threadIdx.x) >> 5;
    int tiles = N >> 4;
    if (wave >= B * tiles) return;           // wave-uniform
    int b  = wave / tiles;
    int mt = wave - b * tiles;
    int li   = lane & 15;
    int half = lane >> 4;
    int kb   = half << 1;                    // K base: 0 or 2
    int row0 = mt << 4;

    const float* At = AT + (size_t)b * 4 * N;
    const float* Bt = BT + (size_t)b * 4 * N;
    const float* cb = bn2 + (size_t)b * N;

    v2f a;
    a.x = At[kb * N + row0 + li];
    a.y = At[(kb + 1) * N + row0 + li];

    float rn2[8];
    #pragma unroll
    for (int j = 0; j < 8; ++j)
        rn2[j] = an2[(size_t)b * N + row0 + (half << 3) + j];

    float dmin[8]; int imin[8];
    #pragma unroll
    for (int j = 0; j < 8; ++j) { dmin[j] = 3.0e38f; imin[j] = 0; }

    int nt = 0;
    // main loop: 4 gc tiles per trip (tiles = N/16 is a multiple of 4 for N%64==0)
    for (; nt + 4 <= tiles; nt += 4) {
        v2f bb[4];
        float cn2[4];
        #pragma unroll
        for (int u = 0; u < 4; ++u) {
            int col = ((nt + u) << 4) + li;
            bb[u].x = Bt[kb * N + col];
            bb[u].y = Bt[(kb + 1) * N + col];
            cn2[u]  = cb[col];
        }
        #pragma unroll
        for (int u = 0; u < 4; ++u) {
            int col = ((nt + u) << 4) + li;
            v8f c = {0.f, 0.f, 0.f, 0.f, 0.f, 0.f, 0.f, 0.f};
            c = __builtin_amdgcn_wmma_f32_16x16x4_f32(
                    /*neg_a=*/false, a, /*neg_b=*/false, bb[u],
                    /*c_mod=*/(short)0, c, /*reuse_a=*/false, /*reuse_b=*/false);
            #pragma unroll
            for (int j = 0; j < 8; ++j) {
                float d = rn2[j] + cn2[u] - 2.0f * c[j];
                if (d < dmin[j]) { dmin[j] = d; imin[j] = col; }
            }
        }
    }
    // tail (not reached for N=4096, kept for generality)
    for (; nt < tiles; ++nt) {
        int col = (nt << 4) + li;
        v2f bb;
        bb.x = Bt[kb * N + col];
        bb.y = Bt[(kb + 1) * N + col];
        v8f c = {0.f, 0.f, 0.f, 0.f, 0.f, 0.f, 0.f, 0.f};
        c = __builtin_amdgcn_wmma_f32_16x16x4_f32(
                false, a, false, bb, (short)0, c, false, false);
        float cn2 = cb[col];
        #pragma unroll
        for (int j = 0; j < 8; ++j) {
            float d = rn2[j] + cn2 - 2.0f * c[j];
            if (d < dmin[j]) { dmin[j] = d; imin[j] = col; }
        }
    }

    // min-reduce across the 16 lanes of each half (columns), keep argmin
    #pragma unroll
    for (int j = 0; j < 8; ++j) {
        #pragma unroll
        for (int off = 8; off >= 1; off >>= 1) {
            float od = __shfl_xor(dmin[j], off, 32);
            int   oi = __shfl_xor(imin[j], off, 32);
            if (od < dmin[j] || (od == dmin[j] && oi < imin[j])) {
                dmin[j] = od; imin[j] = oi;
            }
        }
    }
    if (li == 0) {
        #pragma unroll
        for (int j = 0; j < 8; ++j) {
            int row = row0 + (half << 3) + j;
            dist_out[(size_t)b * N + row] = dmin[j];
            if (nn_out) nn_out[(size_t)b * N + row] = imin[j];
        }
    }
}

// ---------------------------------------------------------------------------
// Kernel 4: deterministic partial sums of dist1, dist2, normal loss.
// Fixed 64 blocks x 256 threads -> partials[0..63], [64..127], [128..191].
// ---------------------------------------------------------------------------
__global__ void reduce_kernel(const float* __restrict__ dist1,
                              const float* __restrict__ dist2,
                              const float* __restrict__ pn,  // [B,N,3]
                              const float* __restrict__ gn,  // [B,N,3]
                              const int* __restrict__ nn,    // [B,N]
                              float* __restrict__ partials,
                              int B, int N) {
    __shared__ float sm0[256], sm1[256], sm2[256];
    int total = B * N;
    float s1 = 0.f, s2 = 0.f, sn = 0.f;
    for (int i = blockIdx.x * blockDim.x + threadIdx.x; i < total;
         i += gridDim.x * blockDim.x) {
        s1 += dist1[i];
        s2 += dist2[i];
        int b = i / N;
        int j = nn[i];
        const float* p = pn + (size_t)i * 3;
        const float* g = gn + ((size_t)b * N + j) * 3;
        float dt = p[0] * g[0] + p[1] * g[1] + p[2] * g[2];
        sn += 1.0f - fabsf(dt);
    }
    int tid = threadIdx.x;
    sm0[tid] = s1; sm1[tid] = s2; sm2[tid] = sn;
    __syncthreads();
    for (int st = 128; st > 0; st >>= 1) {
        if (tid < st) {
            sm0[tid] += sm0[tid + st];
            sm1[tid] += sm1[tid + st];
            sm2[tid] += sm2[tid + st];
        }
        __syncthreads();
    }
    if (tid == 0) {
        partials[blockIdx.x]       = sm0[0];
        partials[64 + blockIdx.x]  = sm1[0];
        partials[128 + blockIdx.x] = sm2[0];
    }
}

// ---------------------------------------------------------------------------
// Kernel 5: edge loss partial sums -> partials[192..255].
// ---------------------------------------------------------------------------
__global__ void edge_kernel(const float* __restrict__ verts,  // [B,V,3]
                            const int* __restrict__ faces,    // [F,3]
                            float* __restrict__ partials,
                            int B, int V, int F) {
    __shared__ float sm[256];
    int total = B * F;
    float acc = 0.f;
    for (int i = blockIdx.x * blockDim.x + threadIdx.x; i < total;
         i += gridDim.x * blockDim.x) {
        int b = i / F, f = i - b * F;
        int i0 = faces[3 * f], i1 = faces[3 * f + 1], i2 = faces[3 * f + 2];
        const float* vb = verts + (size_t)b * V * 3;
        float ax = vb[3 * i0], ay = vb[3 * i0 + 1], az = vb[3 * i0 + 2];
        float bx = vb[3 * i1], by = vb[3 * i1 + 1], bz = vb[3 * i1 + 2];
        float cx = vb[3 * i2], cy = vb[3 * i2 + 1], cz = vb[3 * i2 + 2];
        float dx, dy, dz;
        dx = ax - bx; dy = ay - by; dz = az - bz; acc += dx * dx + dy * dy + dz * dz;
        dx = bx - cx; dy = by - cy; dz = bz - cz; acc += dx * dx + dy * dy + dz * dz;
        dx = cx - ax; dy = cy - ay; dz = cz - az; acc += dx * dx + dy * dy + dz * dz;
    }
    int tid = threadIdx.x;
    sm[tid] = acc;
    __syncthreads();
    for (int st = 128; st > 0; st >>= 1) {
        if (tid < st) sm[tid] += sm[tid + st];
        __syncthreads();
    }
    if (tid == 0) partials[192 + blockIdx.x] = sm[0];
}

// ---------------------------------------------------------------------------
// Kernel 6: deterministic finalize.
// ---------------------------------------------------------------------------
__global__ void finalize_kernel(const float* __restrict__ partials,
                                float* __restrict__ out,
                                int B, int N, int F) {
    if (blockIdx.x == 0 && threadIdx.x == 0) {
        float s1 = 0.f, s2 = 0.f, sn = 0.f, se = 0.f;
        for (int i = 0; i < 64; ++i) {
            s1 += partials[i];
            s2 += partials[64 + i];
            sn += partials[128 + i];
            se += partials[192 + i];
        }
        float bn = (float)B * (float)N;
        float chamfer = s1 / bn + s2 / bn;
        float norml   = sn / bn;
        float edgel   = se / ((float)B * (float)F * 3.0f);
        out[0] = CHAMFER_W * chamfer + NORM_W * norml + EDGE_W * edgel;
    }
}

// ---------------------------------------------------------------------------
extern "C" void kernel_launch(void* const* d_in, const int* in_sizes, int n_in,
                              void* d_out, int out_size, void* d_ws, size_t ws_size,
                              hipStream_t stream) {
    const float* pred_v  = (const float*)d_in[0];  // [B,V,3]
    const int*   pred_f  = (const int*)d_in[1];    // [F,3]
    // d_in[2], d_in[3] (gt mesh) are unused by the reference.
    const float* rnd_p   = (const float*)d_in[4];  // [B,N,3]
    const float* rnd_g   = (const float*)d_in[5];  // [B,N,3]
    float* out = (float*)d_out;

    const int B = 8;
    const int V = in_sizes[0] / (3 * B);
    const int F = in_sizes[1] / 3;
    const int N = in_sizes[4] / (3 * B);

    // workspace layout (floats)
    float* ws = (float*)d_ws;
    size_t off = 0;
    float* normals = ws + off; off += (size_t)B * F * 3;
    float* areas   = ws + off; off += (size_t)B * F;
    float* pcT     = ws + off; off += (size_t)B * 4 * N;
    float* gcT     = ws + off; off += (size_t)B * 4 * N;
    float* pn      = ws + off; off += (size_t)B * N * 3;
    float* gn      = ws + off; off += (size_t)B * N * 3;
    float* pn2     = ws + off; off += (size_t)B * N;
    float* gn2     = ws + off; off += (size_t)B * N;
    float* dist1   = ws + off; off += (size_t)B * N;
    float* dist2   = ws + off; off += (size_t)B * N;
    int*   nn      = (int*)(ws + off); off += (size_t)B * N;
    float* partials = ws + off; off += 256;

    // 1) face normals + areas
    {
        int total = B * F;
        int blocks = (total + 255) / 256;
        face_kernel<<<blocks, 256, 0, stream>>>(pred_v, pred_f, normals, areas,
                                                B, V, F);
    }
    // 2) CDF scan + sampling (one block per batch, CDF lives in LDS)
    {
        size_t smem = (size_t)F * sizeof(float) + 32 * sizeof(float);
        sample_kernel<<<B, 1024, smem, stream>>>(pred_v, pred_f, normals, areas,
                                                 rnd_p, rnd_g, pcT, gcT,
                                                 pn, gn, pn2, gn2, B, V, F, N);
    }
    // 3) fused chamfer via WMMA: dist1 + argmin, then dist2 (roles swapped)
    {
        int waves  = B * (N >> 4);
        int blocks = (waves * 32 + 255) / 256;
        chamfer_kernel<<<blocks, 256, 0, stream>>>(pcT, gcT, pn2, gn2,
                                                   dist1, nn, B, N);
        chamfer_kernel<<<blocks, 256, 0, stream>>>(gcT, pcT, gn2, pn2,
                                                   dist2, (int*)nullptr, B, N);
    }
    // 4) deterministic partial reductions
    reduce_kernel<<<64, 256, 0, stream>>>(dist1, dist2, pn, gn, nn, partials, B, N);
    edge_kernel<<<64, 256, 0, stream>>>(pred_v, pred_f, partials, B, V, F);
    // 5) finalize
    finalize_kernel<<<1, 64, 0, stream>>>(partials, out, B, N, F);
}